// GeodesicVariationalAttention_53669911331146
// MI455X (gfx1250) — compile-verified
//
#include <hip/hip_runtime.h>
#include <math.h>

// ---------------------------------------------------------------------------
// Problem constants (match reference)
// ---------------------------------------------------------------------------
#define BB    4
#define NN    1024
#define DIM   768
#define HH    12
#define DD    64
#define RR    8
#define INNER 768            // H*D
#define ROWS  (BB*NN)        // 4096 rows for the big GEMMs
#define QSZ   (ROWS*INNER)   // 3145728 floats per activation buffer
#define LSZ   (BB*HH*NN*RR)  // 393216 floats for lq / lk
#define SCALEV 0.125f        // D^-0.5

typedef float v2f __attribute__((ext_vector_type(2)));
typedef float v8f __attribute__((ext_vector_type(8)));
typedef unsigned int v4u __attribute__((ext_vector_type(4)));
typedef int v8i __attribute__((ext_vector_type(8)));
typedef int v4i __attribute__((ext_vector_type(4)));

static __device__ __forceinline__ v8f wmma4(v2f a, v2f b, v8f c) {
  // D(16x16,f32) += A(16x4,f32) * B(4x16,f32)   -- full fp32 precision
  return __builtin_amdgcn_wmma_f32_16x16x4_f32(
      /*neg_a=*/false, a, /*neg_b=*/false, b,
      /*c_mod=*/(short)0, c, /*reuse_a=*/false, /*reuse_b=*/false);
}

// ---------------------------------------------------------------------------
// TDM: DMA a 16-row x 64-col fp32 tile (row stride = INNER elements) from
// global memory into LDS.  D# built per CDNA5 ISA §8 (2-D tensor, groups 2/3
// zero).  All descriptor words are wave-uniform (readfirstlane-forced).
// Uses the 6-arg clang-23 builtin form (extra zero-filled int32x8 group).
// ---------------------------------------------------------------------------
static __device__ __forceinline__ void tdm_load_v_tile(const float* gsrc,
                                                       float* lds_dst) {
  unsigned long long ga = (unsigned long long)(uintptr_t)gsrc;
  unsigned int ga_lo  = __builtin_amdgcn_readfirstlane((unsigned int)ga);
  unsigned int ga_hi  = __builtin_amdgcn_readfirstlane((unsigned int)(ga >> 32));
  unsigned int lds_off = __builtin_amdgcn_readfirstlane(
      (unsigned int)(uintptr_t)lds_dst);       // generic->LDS: low 32 bits

  v4u g0;
  g0[0] = 1u;                                  // count=1, user descriptor
  g0[1] = lds_off;                             // lds_addr (bytes)
  g0[2] = ga_lo;                               // global_addr[31:0]
  g0[3] = (ga_hi & 0x01FFFFFFu) | (2u << 30);  // global_addr[56:32] | type=2

  v8i g1;
  g1[0] = (2 << 16);                 // data_size=4B; no multicast/pad/iterate
  g1[1] = (64 << 16);                // tensor_dim0 = 64   (bits 79:48)
  g1[2] = (16 << 16);                // tensor_dim1 = 16   (bits 111:80)
  g1[3] = (64 << 16);                // tile_dim0  = 64    (bits 127:112)
  g1[4] = 16;                        // tile_dim1=16, tile_dim2=0
  g1[5] = INNER;                     // tensor_dim0_stride = 768 (low 32)
  g1[6] = (int)((unsigned)(16 * INNER) << 16);  // tensor_dim1_stride[15:0]
  g1[7] = (int)((unsigned)(16 * INNER) >> 16);  // tensor_dim1_stride[47:16]

  v4i gz4 = {0, 0, 0, 0};
  v8i gz8 = {0, 0, 0, 0, 0, 0, 0, 0};
  __builtin_amdgcn_tensor_load_to_lds(g0, g1, gz4, gz4, gz8, 0);
}

// ---------------------------------------------------------------------------
// Kernel 1: QKV projection.  C(4096x768) = X(4096x768) * W^T, W row-major so
// B-fragment element B[k][n] = W[n][k].  Each wave computes a 16x64 strip
// (4 column tiles) so the A fragment is reused 4x per k-step; the 3 extra
// weight columns are reached through constant instruction offsets.
// ---------------------------------------------------------------------------
__global__ __launch_bounds__(256) void qkv_gemm_kernel(
    const float* __restrict__ x,
    const float* __restrict__ wq, const float* __restrict__ wk,
    const float* __restrict__ wv,
    float* __restrict__ q, float* __restrict__ k, float* __restrict__ v) {
  const float* W;
  float* C;
  if (blockIdx.z == 0)      { W = wq; C = q; }
  else if (blockIdx.z == 1) { W = wk; C = k; }
  else                      { W = wv; C = v; }

  const int wave = threadIdx.x >> 5;
  const int lane = threadIdx.x & 31;
  const int lo   = lane & 15;
  const int hi2  = (lane >> 4) * 2;

  const int tm    = blockIdx.y * 8 + wave;   // 256 row tiles
  const int cbase = blockIdx.x * 64;         // 12 column strips

  const float* xrow = x + (size_t)(tm * 16 + lo) * DIM;   // A row (M = lo)
  const float* wcol = W + (size_t)(cbase + lo) * DIM;     // B cols, +c*16*DIM

  v8f acc0 = {0.f,0.f,0.f,0.f,0.f,0.f,0.f,0.f};
  v8f acc1 = acc0, acc2 = acc0, acc3 = acc0;

  for (int kk0 = 0; kk0 < DIM; kk0 += 32) {
    __builtin_prefetch(xrow + kk0 + 256, 0, 0);
    __builtin_prefetch(wcol + kk0 + 256, 0, 0);
    __builtin_prefetch(wcol + 16 * DIM + kk0 + 256, 0, 0);
    __builtin_prefetch(wcol + 32 * DIM + kk0 + 256, 0, 0);
    __builtin_prefetch(wcol + 48 * DIM + kk0 + 256, 0, 0);
#pragma unroll
    for (int kk = kk0; kk < kk0 + 32; kk += 4) {
      v2f a  = *(const v2f*)(xrow + kk + hi2);
      v2f b0 = *(const v2f*)(wcol + kk + hi2);
      v2f b1 = *(const v2f*)(wcol + 16 * DIM + kk + hi2);
      v2f b2 = *(const v2f*)(wcol + 32 * DIM + kk + hi2);
      v2f b3 = *(const v2f*)(wcol + 48 * DIM + kk + hi2);
      acc0 = wmma4(a, b0, acc0);
      acc1 = wmma4(a, b1, acc1);
      acc2 = wmma4(a, b2, acc2);
      acc3 = wmma4(a, b3, acc3);
    }
  }

  const int rbase = tm * 16 + (lane >> 4) * 8;
  float* crow = C + (size_t)rbase * INNER + cbase + lo;
#pragma unroll
  for (int r = 0; r < 8; ++r) {
    crow[(size_t)r * INNER + 0]  = acc0[r];
    crow[(size_t)r * INNER + 16] = acc1[r];
    crow[(size_t)r * INNER + 32] = acc2[r];
    crow[(size_t)r * INNER + 48] = acc3[r];
  }
}

// ---------------------------------------------------------------------------
// Kernel 2: SPD projection + log:  l = log(softplus(W*qvec + b) + 1e-6 + 1e-8)
// one thread per (b,h,n,r);  blockIdx.z selects q-side / k-side.
// ---------------------------------------------------------------------------
__global__ __launch_bounds__(256) void spd_log_kernel(
    const float* __restrict__ q, const float* __restrict__ k,
    const float* __restrict__ sq_w, const float* __restrict__ sq_b,
    const float* __restrict__ sk_w, const float* __restrict__ sk_b,
    float* __restrict__ lq, float* __restrict__ lk) {
  const int z = blockIdx.z;
  const float* src = z ? k : q;
  const float* w   = z ? sk_w : sq_w;
  const float* bb  = z ? sk_b : sq_b;
  float* dst       = z ? lk : lq;

  const int tid = blockIdx.x * blockDim.x + threadIdx.x;  // < B*H*N*R
  const int r  = tid & (RR - 1);
  const int t  = tid >> 3;              // (b,h,n) flat
  const int n  = t & (NN - 1);
  const int bh = t >> 10;               // b*H + h
  const int h  = bh % HH;
  const int b  = bh / HH;

  const float* qv = src + ((size_t)(b * NN + n)) * INNER + h * DD;
  const float* wr = w + r * DD;
  float s = bb[r];
#pragma unroll 8
  for (int d = 0; d < DD; ++d) s += qv[d] * wr[d];
  float sp = (s > 20.f) ? s : log1pf(__expf(s));  // softplus
  dst[tid] = __logf(sp + 1e-6f + 1e-8f);          // lq/lk index == tid
}

// ---------------------------------------------------------------------------
// Kernel 3: fused geodesic attention (flash style, online softmax).
// One wave per (b,h,16-query tile); 8 waves / block.
//   logits s_ij = -SCALE * sqrt(max(|lq_i|^2 + |lk_j|^2 - 2 lq_i.lk_j, 1e-12))
//   O = softmax_j(s) @ V     accumulated with f32 WMMA.
// V tiles are DMA'd into per-wave LDS slices by the Tensor Data Mover,
// double-buffered and synchronized with s_wait_tensorcnt.
// ---------------------------------------------------------------------------
__global__ __launch_bounds__(256) void attn_kernel(
    const float* __restrict__ lq, const float* __restrict__ lk,
    const float* __restrict__ v, float* __restrict__ o) {
  __shared__ float pbuf[8][16][16];       // per-wave P-tile transpose buffer
  __shared__ float vbuf[8][2][16][64];    // per-wave double-buffered V tiles

  const int wave = threadIdx.x >> 5;
  const int lane = threadIdx.x & 31;
  const int lo   = lane & 15;
  const int hi   = lane >> 4;
  const int hi2  = hi * 2;

  const int tile = blockIdx.x * 8 + wave;  // 0..3071
  const int qt   = tile & 63;              // query tile in [0,64)
  const int bh   = tile >> 6;              // b*H + h
  const int h    = bh % HH;
  const int b    = bh / HH;
  const int q0   = qt * 16;

  const float* lqb = lq + (size_t)bh * NN * RR;
  const float* lkb = lk + (size_t)bh * NN * RR;
  const float* vb  = v + (size_t)b * NN * INNER + h * DD;

  // |lq_i|^2 for the 8 rows this lane's accumulator components map to
  float sqi[8];
#pragma unroll
  for (int r = 0; r < 8; ++r) {
    const float* lr = lqb + (size_t)(q0 + r + 8 * hi) * RR;
    float s = 0.f;
#pragma unroll
    for (int t = 0; t < RR; ++t) s += lr[t] * lr[t];
    sqi[r] = s;
  }
  // A-fragments of lq tile (16x8), reused across all key tiles
  const float* lqm = lqb + (size_t)(q0 + lo) * RR;
  v2f aq0 = *(const v2f*)(lqm + hi2);
  v2f aq1 = *(const v2f*)(lqm + 4 + hi2);

  float mrun[8], lrun[8];
  v8f   O[4];
#pragma unroll
  for (int r = 0; r < 8; ++r) { mrun[r] = -3.0e38f; lrun[r] = 0.f; }
#pragma unroll
  for (int dt = 0; dt < 4; ++dt) O[dt] = (v8f){0.f,0.f,0.f,0.f,0.f,0.f,0.f,0.f};

  float (*P)[16] = pbuf[wave];

  // prime the V-tile pipeline
  tdm_load_v_tile(vb, &vbuf[wave][0][0][0]);

  for (int t = 0; t < NN / 16; ++t) {
    const int j0 = t * 16;
    if (t + 1 < NN / 16) {
      tdm_load_v_tile(vb + (size_t)(j0 + 16) * INNER,
                      &vbuf[wave][(t + 1) & 1][0][0]);
      __builtin_amdgcn_s_wait_tensorcnt(1);   // current tile landed
    } else {
      __builtin_amdgcn_s_wait_tensorcnt(0);
    }
    const float (*VT)[64] = (const float(*)[64])&vbuf[wave][t & 1][0][0];

    const int jn = j0 + lo;
    const float* lkr = lkb + (size_t)jn * RR;
    float skj = 0.f;
#pragma unroll
    for (int tt = 0; tt < RR; ++tt) skj += lkr[tt] * lkr[tt];

    // cross = lq_tile(16x8) * lk_tile^T(8x16): B[k][n] = lk[jn][k]
    v2f bk0 = *(const v2f*)(lkr + hi2);
    v2f bk1 = *(const v2f*)(lkr + 4 + hi2);
    v8f cr = {0.f,0.f,0.f,0.f,0.f,0.f,0.f,0.f};
    cr = wmma4(aq0, bk0, cr);
    cr = wmma4(aq1, bk1, cr);

    // logits + online softmax (row reductions over 16-lane halves)
#pragma unroll
    for (int r = 0; r < 8; ++r) {
      float d2 = sqi[r] + skj - 2.f * cr[r];
      float s  = -SCALEV * sqrtf(fmaxf(d2, 1e-12f));
      float mx = s;
#pragma unroll
      for (int ofs = 1; ofs < 16; ofs <<= 1) mx = fmaxf(mx, __shfl_xor(mx, ofs, 32));
      float newm  = fmaxf(mrun[r], mx);
      float pe    = __expf(s - newm);
      float sm    = pe;
#pragma unroll
      for (int ofs = 1; ofs < 16; ofs <<= 1) sm += __shfl_xor(sm, ofs, 32);
      float alpha = __expf(mrun[r] - newm);
      mrun[r] = newm;
      lrun[r] = lrun[r] * alpha + sm;
#pragma unroll
      for (int dt = 0; dt < 4; ++dt) O[dt][r] *= alpha;
      P[r + 8 * hi][lo] = pe;   // stage P tile (C-layout -> LDS)
    }
    asm volatile("" ::: "memory");  // same-wave DS ops are in-order; block reordering

    // O += P(16x16) * V_tile(16x64), K split into 4 WMMA steps; V from LDS
#pragma unroll
    for (int ks = 0; ks < 4; ++ks) {
      const int kc = ks * 4 + hi2;
      v2f ap = { P[lo][kc], P[lo][kc + 1] };   // A-fragment of P from LDS
#pragma unroll
      for (int dt = 0; dt < 4; ++dt) {
        const int dcol = dt * 16 + lo;
        v2f bv = { VT[kc][dcol], VT[kc + 1][dcol] };  // B[k][d] = V[j0+k][d]
        O[dt] = wmma4(ap, bv, O[dt]);
      }
    }
  }

  // normalize and write out (B,N,INNER)
  float* ob = o + (size_t)b * NN * INNER + h * DD;
#pragma unroll
  for (int dt = 0; dt < 4; ++dt) {
#pragma unroll
    for (int r = 0; r < 8; ++r) {
      const int row = q0 + r + 8 * hi;
      ob[(size_t)row * INNER + dt * 16 + lo] = O[dt][r] / lrun[r];
    }
  }
}

// ---------------------------------------------------------------------------
// Kernel 4: output projection  out = o * wo^T + bias  (writes d_out[0..QSZ))
// Same 16x64-strip-per-wave structure as kernel 1.
// ---------------------------------------------------------------------------
__global__ __launch_bounds__(256) void out_gemm_kernel(
    const float* __restrict__ o, const float* __restrict__ wo,
    const float* __restrict__ bias, float* __restrict__ out) {
  const int wave = threadIdx.x >> 5;
  const int lane = threadIdx.x & 31;
  const int lo   = lane & 15;
  const int hi2  = (lane >> 4) * 2;

  const int tm    = blockIdx.y * 8 + wave;
  const int cbase = blockIdx.x * 64;

  const float* orow = o  + (size_t)(tm * 16 + lo) * INNER;
  const float* wcol = wo + (size_t)(cbase + lo) * INNER;

  v8f acc0 = {0.f,0.f,0.f,0.f,0.f,0.f,0.f,0.f};
  v8f acc1 = acc0, acc2 = acc0, acc3 = acc0;

  for (int kk0 = 0; kk0 < INNER; kk0 += 32) {
    __builtin_prefetch(orow + kk0 + 256, 0, 0);
    __builtin_prefetch(wcol + kk0 + 256, 0, 0);
    __builtin_prefetch(wcol + 16 * INNER + kk0 + 256, 0, 0);
    __builtin_prefetch(wcol + 32 * INNER + kk0 + 256, 0, 0);
    __builtin_prefetch(wcol + 48 * INNER + kk0 + 256, 0, 0);
#pragma unroll
    for (int kk = kk0; kk < kk0 + 32; kk += 4) {
      v2f a  = *(const v2f*)(orow + kk + hi2);
      v2f b0 = *(const v2f*)(wcol + kk + hi2);
      v2f b1 = *(const v2f*)(wcol + 16 * INNER + kk + hi2);
      v2f b2 = *(const v2f*)(wcol + 32 * INNER + kk + hi2);
      v2f b3 = *(const v2f*)(wcol + 48 * INNER + kk + hi2);
      acc0 = wmma4(a, b0, acc0);
      acc1 = wmma4(a, b1, acc1);
      acc2 = wmma4(a, b2, acc2);
      acc3 = wmma4(a, b3, acc3);
    }
  }

  const int rbase = tm * 16 + (lane >> 4) * 8;
  float* crow = out + (size_t)rbase * DIM + cbase + lo;
  const float bv0 = bias[cbase + lo];
  const float bv1 = bias[cbase + 16 + lo];
  const float bv2 = bias[cbase + 32 + lo];
  const float bv3 = bias[cbase + 48 + lo];
#pragma unroll
  for (int r = 0; r < 8; ++r) {
    crow[(size_t)r * DIM + 0]  = acc0[r] + bv0;
    crow[(size_t)r * DIM + 16] = acc1[r] + bv1;
    crow[(size_t)r * DIM + 32] = acc2[r] + bv2;
    crow[(size_t)r * DIM + 48] = acc3[r] + bv3;
  }
}

// ---------------------------------------------------------------------------
// Kernels 5/6: deterministic KL reduction.
// kl = 0.5 * sum over 4 weight tensors of (exp(2ls) + mu^2 - 1 - 2ls)
// ---------------------------------------------------------------------------
__global__ __launch_bounds__(256) void kl_partial_kernel(
    const float* __restrict__ mu0, const float* __restrict__ ls0,
    const float* __restrict__ mu1, const float* __restrict__ ls1,
    const float* __restrict__ mu2, const float* __restrict__ ls2,
    const float* __restrict__ mu3, const float* __restrict__ ls3,
    float* __restrict__ partial) {
  __shared__ float red[256];
  const int z = blockIdx.z;
  const float* mu = (z == 0) ? mu0 : (z == 1) ? mu1 : (z == 2) ? mu2 : mu3;
  const float* ls = (z == 0) ? ls0 : (z == 1) ? ls1 : (z == 2) ? ls2 : ls3;
  const int nelem = INNER * DIM;  // 589824 for every weight tensor here

  float s = 0.f;
  for (int i = blockIdx.x * 256 + threadIdx.x; i < nelem; i += 256 * 256) {
    float m = mu[i], l = ls[i];
    s += __expf(2.f * l) + m * m - 1.f - 2.f * l;
  }
  red[threadIdx.x] = s;
  __syncthreads();
  for (int st = 128; st > 0; st >>= 1) {
    if (threadIdx.x < st) red[threadIdx.x] += red[threadIdx.x + st];
    __syncthreads();
  }
  if (threadIdx.x == 0) partial[z * 256 + blockIdx.x] = red[0];
}

__global__ __launch_bounds__(1024) void kl_final_kernel(
    const float* __restrict__ partial, float* __restrict__ dst) {
  __shared__ float red[1024];
  red[threadIdx.x] = partial[threadIdx.x];
  __syncthreads();
  for (int st = 512; st > 0; st >>= 1) {
    if (threadIdx.x < st) red[threadIdx.x] += red[threadIdx.x + st];
    __syncthreads();
  }
  if (threadIdx.x == 0) dst[0] = 0.5f * red[0];
}

// ---------------------------------------------------------------------------
// Launcher
// ---------------------------------------------------------------------------
extern "C" void kernel_launch(void* const* d_in, const int* in_sizes, int n_in,
                              void* d_out, int out_size, void* d_ws, size_t ws_size,
                              hipStream_t stream) {
  const float* x     = (const float*)d_in[0];
  const float* wq_mu = (const float*)d_in[1];
  const float* wq_ls = (const float*)d_in[2];
  const float* wk_mu = (const float*)d_in[3];
  const float* wk_ls = (const float*)d_in[4];
  const float* wv_mu = (const float*)d_in[5];
  const float* wv_ls = (const float*)d_in[6];
  const float* wo_mu = (const float*)d_in[7];
  const float* wo_ls = (const float*)d_in[8];
  const float* wo_b  = (const float*)d_in[9];
  const float* sq_w  = (const float*)d_in[10];
  const float* sq_b  = (const float*)d_in[11];
  const float* sk_w  = (const float*)d_in[12];
  const float* sk_b  = (const float*)d_in[13];

  float* out = (float*)d_out;             // [QSZ] activations + [1] kl
  float* ws  = (float*)d_ws;
  float* q   = ws;                        // QSZ
  float* k   = ws + (size_t)QSZ;          // QSZ
  float* v   = ws + (size_t)2 * QSZ;      // QSZ
  float* o   = ws + (size_t)3 * QSZ;      // QSZ
  float* lq  = ws + (size_t)4 * QSZ;      // LSZ
  float* lk  = lq + (size_t)LSZ;          // LSZ
  float* prt = lk + (size_t)LSZ;          // 1024

  // 1) QKV projections (f32 WMMA GEMM, 16x64 strip per wave)
  qkv_gemm_kernel<<<dim3(DIM / 64, ROWS / 128, 3), 256, 0, stream>>>(
      x, wq_mu, wk_mu, wv_mu, q, k, v);

  // 2) SPD projection + log
  spd_log_kernel<<<dim3((BB * HH * NN * RR) / 256, 1, 2), 256, 0, stream>>>(
      q, k, sq_w, sq_b, sk_w, sk_b, lq, lk);

  // 3) fused geodesic attention (TDM-staged V tiles)
  attn_kernel<<<dim3((BB * HH * (NN / 16)) / 8), 256, 0, stream>>>(lq, lk, v, o);

  // 4) output projection + bias -> d_out
  out_gemm_kernel<<<dim3(DIM / 64, ROWS / 128), 256, 0, stream>>>(o, wo_mu, wo_b, out);

  // 5/6) deterministic KL reduction -> d_out[QSZ]
  kl_partial_kernel<<<dim3(256, 1, 4), 256, 0, stream>>>(
      wq_mu, wq_ls, wk_mu, wk_ls, wv_mu, wv_ls, wo_mu, wo_ls, prt);
  kl_final_kernel<<<1, 1024, 0, stream>>>(prt, out + (size_t)QSZ);
}